// QuasiNewtonInitWithExact_44899588112968
// MI455X (gfx1250) — compile-verified
//
#include <hip/hip_runtime.h>

// Batched symmetric eigen-pseudoinverse for 16384 x (64x64) fp32 matrices.
// One workgroup (64 threads = 2 wave32) per matrix:
//   1) coalesced load + in-LDS symmetrization
//   2) parallel cyclic (tournament-ordered) Jacobi eigen-iteration in LDS
//   3) eigenvalue magnitude inversion with tol = 1e-6
//   4) out = V * diag(einv) * V^T reconstructed with V_WMMA_F32_16X16X4_F32
// LDS row stride 65 (odd) -> bank-conflict-free for BOTH row and column walks.

typedef __attribute__((ext_vector_type(2))) float v2f;
typedef __attribute__((ext_vector_type(8))) float v8f;

#define DD 64
#define LS 65           // LDS row stride in floats
#define NSWEEPS 8
#define TOL_INV_EIG 1e-6f

__global__ __launch_bounds__(64)
void jacobi_pinv64_kernel(const float* __restrict__ hes,
                          float* __restrict__ out, int nb)
{
    __shared__ float A[DD * LS];      // working matrix, later W = V*diag(einv)
    __shared__ float V[DD * LS];      // accumulated eigenvectors
    __shared__ float c_of[DD];        // per-column cosine
    __shared__ float s_of[DD];        // per-column signed sine (s at p, -s at q)
    __shared__ int   pairP[DD / 2];
    __shared__ int   pairQ[DD / 2];
    __shared__ int   perm[DD];        // tournament ordering state
    __shared__ float einv[DD];

    const int tid  = threadIdx.x;     // 0..63
    const int lane = tid & 31;
    const int wave = tid >> 5;
    const int b    = blockIdx.x;
    if (b >= nb) return;              // uniform per block (grid == nb)

    const float* __restrict__ Ain = hes + (size_t)b * DD * DD;
    float*       __restrict__ Out = out + (size_t)b * DD * DD;

    // ---------- load (coalesced: threads sweep a row) + identity V ----------
    for (int r = 0; r < DD; ++r) {
        A[r * LS + tid] = Ain[r * DD + tid];
        V[r * LS + tid] = (r == tid) ? 1.0f : 0.0f;
    }
    perm[tid] = tid;
    __syncthreads();

    // ---------- symmetrize in LDS: A <- 0.5*(A + A^T) ----------
    // element (i,j) with i<j is owned by thread i; race-free (see write pattern)
    for (int j = tid + 1; j < DD; ++j) {
        float s = 0.5f * (A[tid * LS + j] + A[j * LS + tid]);
        A[tid * LS + j] = s;
        A[j * LS + tid] = s;
    }
    __syncthreads();

    // ---------- parallel cyclic Jacobi ----------
    for (int sweep = 0; sweep < NSWEEPS; ++sweep) {
        for (int round = 0; round < DD - 1; ++round) {
            // 32 disjoint pivots this round: (perm[i], perm[i+32])
            if (tid < DD / 2) {
                int p = perm[tid], q = perm[tid + DD / 2];
                if (p > q) { int t = p; p = q; q = t; }
                float app = A[p * LS + p];
                float aqq = A[q * LS + q];
                float apq = A[p * LS + q];
                float c = 1.0f, s = 0.0f;
                if (__builtin_fabsf(apq) > 1e-20f) {
                    float theta = (aqq - app) / (2.0f * apq);
                    float t = 1.0f / (__builtin_fabsf(theta) +
                                      __builtin_sqrtf(theta * theta + 1.0f));
                    t = (theta < 0.0f) ? -t : t;
                    c = 1.0f / __builtin_sqrtf(t * t + 1.0f);
                    s = t * c;
                }
                pairP[tid] = p; pairQ[tid] = q;
                c_of[p] = c;  c_of[q] = c;
                s_of[p] = s;  s_of[q] = -s;
            }
            __syncthreads();

            // column phase: A <- A*G, V <- V*G   (thread = row)
            {
                const int row = tid;
                for (int i = 0; i < DD / 2; ++i) {
                    int p = pairP[i], q = pairQ[i];
                    float c = c_of[p], s = s_of[p];
                    float t1 = A[row * LS + p], t2 = A[row * LS + q];
                    A[row * LS + p] = c * t1 - s * t2;
                    A[row * LS + q] = s * t1 + c * t2;
                    float u1 = V[row * LS + p], u2 = V[row * LS + q];
                    V[row * LS + p] = c * u1 - s * u2;
                    V[row * LS + q] = s * u1 + c * u2;
                }
            }
            __syncthreads();

            // row phase: A <- G^T*A   (thread = column)
            {
                const int col = tid;
                for (int i = 0; i < DD / 2; ++i) {
                    int p = pairP[i], q = pairQ[i];
                    float c = c_of[p], s = s_of[p];
                    float t1 = A[p * LS + col], t2 = A[q * LS + col];
                    A[p * LS + col] = c * t1 - s * t2;
                    A[q * LS + col] = s * t1 + c * t2;
                }
            }
            // round-robin tournament rotation of perm (read old, write after barrier)
            int newp;
            if      (tid == 0)          newp = perm[0];
            else if (tid == 1)          newp = perm[DD / 2];
            else if (tid < DD / 2)      newp = perm[tid - 1];
            else if (tid < DD - 1)      newp = perm[tid + 1];
            else                        newp = perm[DD / 2 - 1];
            __syncthreads();
            perm[tid] = newp;
            __syncthreads();
        }
    }

    // ---------- eigenvalue inversion: einv[j] = |d_j|>tol ? 1/|d_j| : 0 ----------
    {
        float d  = A[tid * LS + tid];
        float ad = __builtin_fabsf(d);
        einv[tid] = (ad > TOL_INV_EIG) ? (1.0f / ad) : 0.0f;
    }
    __syncthreads();

    // ---------- W = V * diag(einv), stored over A (thread = column) ----------
    for (int r = 0; r < DD; ++r)
        A[r * LS + tid] = V[r * LS + tid] * einv[tid];
    __syncthreads();

    // ---------- out = W * V^T via V_WMMA_F32_16X16X4_F32 ----------
    // 16 output tiles of 16x16; each wave32 computes 8 tiles, K accumulated
    // in 16 chunks of 4. EXEC is all-1s here (no lane divergence).
    // A-frag layout (32-bit A 16x4): lanes 0-15 hold K=k,k+1 in v0,v1;
    // lanes 16-31 hold K=k+2,k+3. B-frag mirrors it (B[k][n] = V[n][k]).
    const int halfsel = lane >> 4;        // 0: lanes 0-15, 1: lanes 16-31
    const int l16     = lane & 15;
    for (int t = wave * 8; t < wave * 8 + 8; ++t) {
        const int ti = t >> 2, tj = t & 3;
        v8f acc = {0.f, 0.f, 0.f, 0.f, 0.f, 0.f, 0.f, 0.f};
        const int rowA = ti * 16 + l16;   // M index for A-frag
        const int colB = tj * 16 + l16;   // N index for B-frag (row of V)
        for (int kc = 0; kc < DD; kc += 4) {
            const int k0 = kc + 2 * halfsel;
            v2f a, bf;
            a.x  = A[rowA * LS + k0];
            a.y  = A[rowA * LS + k0 + 1];
            bf.x = V[colB * LS + k0];
            bf.y = V[colB * LS + k0 + 1];
            acc = __builtin_amdgcn_wmma_f32_16x16x4_f32(
                      /*neg_a=*/false, a, /*neg_b=*/false, bf,
                      /*c_mod=*/(short)0, acc,
                      /*reuse_a=*/false, /*reuse_b=*/false);
        }
        // D layout: lanes 0-15 -> M = vgpr r, lanes 16-31 -> M = r+8; N = lane%16
        const int mBase = ti * 16 + halfsel * 8;
        const int n     = tj * 16 + l16;
        #pragma unroll
        for (int r = 0; r < 8; ++r)
            Out[(size_t)(mBase + r) * DD + n] = acc[r];
    }
}

extern "C" void kernel_launch(void* const* d_in, const int* in_sizes, int n_in,
                              void* d_out, int out_size, void* d_ws, size_t ws_size,
                              hipStream_t stream) {
    (void)n_in; (void)out_size; (void)d_ws; (void)ws_size;
    const float* hes = (const float*)d_in[0];
    float* out = (float*)d_out;
    const int nb = in_sizes[0] / (DD * DD);   // 16384
    jacobi_pinv64_kernel<<<dim3(nb), dim3(64), 0, stream>>>(hes, out, nb);
}